// ROIPooler_3642132267627
// MI455X (gfx1250) — compile-verified
//
#include <hip/hip_runtime.h>
#include <math.h>

// ---------------------------------------------------------------------------
// ROI-Align (FPN pooler) for MI455X / gfx1250, wave32.
//   inputs : x2 [2,256,200,304], x3 [2,256,100,152], x4 [2,256,50,76],
//            x5 [2,256,25,38], boxes [2,256,4]  (all float32)
//   output : [512, 256, 7, 7] float32
// One block (4 waves) per ROI. Gathers + bilinear staged into LDS; the
// SRxSR sample->bin averaging runs on the matrix pipe via
// V_WMMA_F32_16X16X4_F32 (selector-matrix formulation).
// ---------------------------------------------------------------------------

typedef float v2f __attribute__((ext_vector_type(2)));
typedef float v8f __attribute__((ext_vector_type(8)));

#define ROI_OUT   7
#define NSAMP     14            // OUT * SR
#define NS2       196           // 14*14 sample grid
#define SPITCH    17            // 16 channels + 1 pad (bank-conflict free)
#define NWAVE     4
#define CHAN      256
#define NBINS     49            // 7*7
#define CTILES    16            // 256 / 16
#define CT_PER_WAVE (CTILES / NWAVE)

__global__ __launch_bounds__(128)
void roi_pooler_kernel(const float* __restrict__ x2,
                       const float* __restrict__ x3,
                       const float* __restrict__ x4,
                       const float* __restrict__ x5,
                       const float* __restrict__ boxes,
                       float* __restrict__ out,
                       int n_per_batch)
{
    __shared__ float S[NWAVE][NS2 * SPITCH];   // per-wave sample values
    __shared__ int   sY0[NSAMP], sY1[NSAMP], sX0[NSAMP], sX1[NSAMP];
    __shared__ float sFY[NSAMP], sFX[NSAMP];
    __shared__ int   sVY[NSAMP], sVX[NSAMP];

    const int r = blockIdx.x;                 // ROI index, 0..511
    const int b = r / n_per_batch;            // batch index

    // -------- uniform per-ROI parameters (every thread, identical) --------
    const float X1 = boxes[(size_t)r * 4 + 0];
    const float Y1 = boxes[(size_t)r * 4 + 1];
    const float X2 = boxes[(size_t)r * 4 + 2];
    const float Y2 = boxes[(size_t)r * 4 + 3];

    float roi_size = sqrtf((X2 - X1) * (Y2 - Y1));
    float lvf = floorf(4.0f + log2f(roi_size / 224.0f + 1e-8f));
    lvf = fminf(fmaxf(lvf, 2.0f), 5.0f);
    const int lvl = (int)lvf - 2;             // 0..3

    const float scales[4] = {0.25f, 0.125f, 0.0625f, 0.03125f};
    const int   Hs[4]     = {200, 100, 50, 25};
    const int   Ws[4]     = {304, 152, 76, 38};
    const float scale = scales[lvl];
    const int   H = Hs[lvl];
    const int   W = Ws[lvl];
    const float* feat = (lvl == 0) ? x2 : (lvl == 1) ? x3 : (lvl == 2) ? x4 : x5;

    const float x1s = X1 * scale - 0.5f;
    const float y1s = Y1 * scale - 0.5f;
    const float bw  = (X2 - X1) * scale * (1.0f / ROI_OUT);
    const float bh  = (Y2 - Y1) * scale * (1.0f / ROI_OUT);

    // -------- sample-grid metadata (14 y's, 14 x's); grid(g)=0.5g+0.25 ----
    if (threadIdx.x < 2 * NSAMP) {
        const int  i   = threadIdx.x;
        const bool isX = (i >= NSAMP);
        const int  g   = isX ? (i - NSAMP) : i;
        const float gv = 0.5f * (float)g + 0.25f;
        const float coord = isX ? (x1s + gv * bw) : (y1s + gv * bh);
        const int   sz    = isX ? W : H;
        const int valid = (coord >= -1.0f) && (coord <= (float)sz);
        float cc  = fminf(fmaxf(coord, 0.0f), (float)(sz - 1));
        float c0f = floorf(cc);
        int   i0  = (int)c0f;
        int   i1  = min(i0 + 1, sz - 1);
        float f   = cc - c0f;
        if (isX) { sX0[g] = i0; sX1[g] = i1; sFX[g] = f; sVX[g] = valid; }
        else     { sY0[g] = i0; sY1[g] = i1; sFY[g] = f; sVY[g] = valid; }
    }
    __syncthreads();

    const int wid  = threadIdx.x >> 5;
    const int lane = threadIdx.x & 31;
    float* Sw = S[wid];

    const size_t bbase = (size_t)b * CHAN * (size_t)H * (size_t)W;
    const size_t plane = (size_t)H * (size_t)W;

    const int n  = lane & 15;       // B/D column (channel within tile), A row M
    const int hi = lane >> 4;       // lane half: selects sy (K split) / M+8

    for (int ct = 0; ct < CT_PER_WAVE; ++ct) {
        const int ctile = wid + ct * NWAVE;   // 0..15
        const int c0    = ctile * 16;

        // ---------------- fill phase: gather + bilinear -> LDS ------------
        for (int cl = 0; cl < 16; ++cl) {
            const float* fp = feat + bbase + (size_t)(c0 + cl) * plane;
            for (int s = lane; s < NS2; s += 32) {
                const int gy = s / NSAMP;
                const int gx = s - gy * NSAMP;   // x fastest -> coalesced-ish
                float v = 0.0f;
                if (sVY[gy] && sVX[gx]) {
                    const int   y0 = sY0[gy], y1 = sY1[gy];
                    const int   xa = sX0[gx], xb = sX1[gx];
                    const float fy = sFY[gy], fx = sFX[gx];
                    const float f00 = fp[(size_t)y0 * W + xa];
                    const float f01 = fp[(size_t)y0 * W + xb];
                    const float f10 = fp[(size_t)y1 * W + xa];
                    const float f11 = fp[(size_t)y1 * W + xb];
                    const float gy1 = 1.0f - fy, gx1 = 1.0f - fx;
                    v = f00 * gy1 * gx1 + f01 * gy1 * fx
                      + f10 * fy  * gx1 + f11 * fy  * fx;
                }
                Sw[s * SPITCH + cl] = v;
            }
        }
        __syncthreads();   // all waves run identical trip counts -> convergent

        // ------------- reduce phase: samples -> bins on matrix pipe -------
        // out[bin, ch] = sum_{k=0..63} A[bin,k] * B[k,ch],
        //   A[bin, 4*bin + s] = 0.25 (selector), B[4t+s, ch] = sample value
        //   of bin t, sub-sample s (s = 2*sy + sx).
        // A 16x4 f32 layout : M = lane&15, VGPR0={K0,K2}, VGPR1={K1,K3}
        // B 4x16  f32 layout: N = lane&15, VGPR0={K0,K2}, VGPR1={K1,K3}
        for (int mt = 0; mt < 4; ++mt) {
            v8f acc = {0.f, 0.f, 0.f, 0.f, 0.f, 0.f, 0.f, 0.f};
            for (int t = 0; t < 16; ++t) {
                const int bin = mt * 16 + t;
                if (bin >= NBINS) break;            // uniform branch
                const float w = (n == t) ? 0.25f : 0.0f;
                v2f a = {w, w};
                const int ph = bin / ROI_OUT;
                const int pw = bin - ph * ROI_OUT;
                // lane half 'hi' selects sy (K=0,1 vs K=2,3)
                const int row = (2 * ph + hi) * NSAMP + 2 * pw;
                v2f bmat = { Sw[row * SPITCH + n],          // sx = 0
                             Sw[(row + 1) * SPITCH + n] };  // sx = 1
                acc = __builtin_amdgcn_wmma_f32_16x16x4_f32(
                        false, a, false, bmat, (short)0, acc, false, false);
            }
            // D layout: VGPR v -> M = v + 8*hi, N = lane&15
#pragma unroll
            for (int v = 0; v < 8; ++v) {
                const int m   = v + hi * 8;
                const int bin = mt * 16 + m;
                if (bin < NBINS) {
                    const int c = c0 + n;
                    out[((size_t)r * CHAN + c) * NBINS + bin] = acc[v];
                }
            }
        }
        __syncthreads();
    }
}

// ---------------------------------------------------------------------------
extern "C" void kernel_launch(void* const* d_in, const int* in_sizes, int n_in,
                              void* d_out, int out_size, void* d_ws, size_t ws_size,
                              hipStream_t stream)
{
    const float* x2    = (const float*)d_in[0];
    const float* x3    = (const float*)d_in[1];
    const float* x4    = (const float*)d_in[2];
    const float* x5    = (const float*)d_in[3];
    const float* boxes = (const float*)d_in[4];
    float*       out   = (float*)d_out;

    const int n_rois      = in_sizes[4] / 4;   // B*N = 512
    const int n_per_batch = n_rois / 2;        // B = 2

    roi_pooler_kernel<<<n_rois, 128, 0, stream>>>(
        x2, x3, x4, x5, boxes, out, n_per_batch);
}